// GCNN_14233521619311
// MI455X (gfx1250) — compile-verified
//
#include <hip/hip_runtime.h>

typedef __attribute__((ext_vector_type(2))) float v2f;
typedef __attribute__((ext_vector_type(8))) float v8f;

// ---------------------------------------------------------------- utilities
__global__ void zero_f32(float* __restrict__ p, long long n) {
    long long i = (long long)blockIdx.x * blockDim.x + threadIdx.x;
    if (i < n) p[i] = 0.0f;
}

__global__ void count_deg(const int* __restrict__ src, const int* __restrict__ dst,
                          float* __restrict__ cnt_out, float* __restrict__ cnt_in, int E) {
    int e = blockIdx.x * blockDim.x + threadIdx.x;
    if (e >= E) return;
    atomicAdd(&cnt_out[src[e]], 1.0f);
    atomicAdd(&cnt_in[dst[e]], 1.0f);
}

__global__ void finalize_deg(float* __restrict__ p, long long n) {
    long long i = (long long)blockIdx.x * blockDim.x + threadIdx.x;
    if (i < n) p[i] = rsqrtf(fmaxf(p[i], 1.0f));
}

// ---------------------------------------------------------------- GEMM (WMMA fp32)
// Y[N x NCOLS] = diag(scale?) * X[N x 128] @ W[128 x NCOLS]
// block: 16 rows; one wave per 16-column panel -> blockDim = 32 * NCOLS/16
template <int NCOLS, bool SCALE>
__global__ __launch_bounds__(256)
void gemm_wmma_f32(const float* __restrict__ X, const float* __restrict__ scale,
                   const float* __restrict__ W, float* __restrict__ Y, int N) {
    __shared__ float Ash[16][128 + 4];
    const int tid = threadIdx.x;
    const int nthreads = 32 * (NCOLS / 16);
    const int rowbase = blockIdx.x * 16;

    // stage scaled A tile (16 x 128) into LDS
    for (int i = tid; i < 16 * 128; i += nthreads) {
        int r = i >> 7, k = i & 127;
        int row = rowbase + r;
        float v = 0.0f;
        if (row < N) {
            v = X[(long long)row * 128 + k];
            if (SCALE) v *= scale[row];
        }
        Ash[r][k] = v;
    }
    __syncthreads();

    const int wave = tid >> 5;
    const int lane = tid & 31;
    const int colbase = wave * 16;
    const int l = lane & 15;            // M (A) / N (B,C) within half
    const int khalf = (lane >> 4) << 1; // K offset 0 or 2 for upper lanes

    // base pointer for B fragments: immediate offsets per k-step (NCOLS const)
    const float* Wp = W + (long long)khalf * NCOLS + colbase + l;

    v8f c = {};
#pragma unroll 8
    for (int kc = 0; kc < 128; kc += 4) {
        v2f a, b;
        a.x = Ash[l][kc + khalf];
        a.y = Ash[l][kc + khalf + 1];
        b.x = Wp[kc * NCOLS];
        b.y = Wp[kc * NCOLS + NCOLS];
        // D = A(16x4,f32) x B(4x16,f32) + C(16x16,f32)
        c = __builtin_amdgcn_wmma_f32_16x16x4_f32(false, a, false, b,
                                                  (short)0, c, false, false);
    }

    // C/D layout: VGPR r -> lanes0-15: M=r, lanes16-31: M=8+r; N = lane&15
    const int mbase = (lane >> 4) << 3;
    float* Yp = Y + (long long)(rowbase + mbase) * NCOLS + colbase + l;
    if (rowbase + 16 <= N) {
#pragma unroll
        for (int r = 0; r < 8; ++r) Yp[r * NCOLS] = c[r];
    } else {
#pragma unroll
        for (int r = 0; r < 8; ++r)
            if (rowbase + mbase + r < N) Yp[r * NCOLS] = c[r];
    }
}

// ---------------------------------------------------------------- edge scatter-add
// agg[dst[e], :] += Y[src[e], :]   (one thread per (edge, 4-channel group))
template <int C>
__global__ void scatter_add(const float* __restrict__ Y, const int* __restrict__ src,
                            const int* __restrict__ dst, float* __restrict__ agg, int E) {
    long long gid = (long long)blockIdx.x * blockDim.x + threadIdx.x;
    const int groups = C >> 2;
    long long total = (long long)E * groups;
    if (gid >= total) return;
    int e = (int)(gid / groups);
    int g = (int)(gid - (long long)e * groups);
    int s = src[e], d = dst[e];
    const float4 v = *(const float4*)(Y + (long long)s * C + g * 4);
    float* p = agg + (long long)d * C + g * 4;
    atomicAdd(p + 0, v.x);
    atomicAdd(p + 1, v.y);
    atomicAdd(p + 2, v.z);
    atomicAdd(p + 3, v.w);
}

// ---------------------------------------------------------------- BatchNorm
// h1 = agg1 * rs_in + b1 ; accumulate per-channel sum / sumsq
__global__ void bn_stats(const float* __restrict__ agg, const float* __restrict__ rs_in,
                         const float* __restrict__ b1, float* __restrict__ stats, int N) {
    int c = threadIdx.x;                 // 128 threads = 128 channels
    float bc = b1[c];
    float s = 0.0f, s2 = 0.0f;
    for (int r = blockIdx.x; r < N; r += gridDim.x) {
        float v = agg[(long long)r * 128 + c] * rs_in[r] + bc;
        s += v;
        s2 += v * v;
    }
    atomicAdd(&stats[c], s);
    atomicAdd(&stats[128 + c], s2);
}

// in-place: buf = relu(bn(buf*rs_in + b1)) * rs_out   (produces X2 for layer 2)
__global__ void bn_apply(float* __restrict__ buf, const float* __restrict__ rs_in,
                         const float* __restrict__ rs_out, const float* __restrict__ b1,
                         const float* __restrict__ gamma, const float* __restrict__ beta,
                         const float* __restrict__ stats, int N) {
    long long gid = (long long)blockIdx.x * blockDim.x + threadIdx.x;
    if (gid >= (long long)N * 128) return;
    int c = (int)(gid & 127);
    int r = (int)(gid >> 7);
    float invN = 1.0f / (float)N;
    float mean = stats[c] * invN;
    float var = stats[128 + c] * invN - mean * mean;
    float v = buf[gid] * rs_in[r] + b1[c];
    v = (v - mean) * rsqrtf(var + 1e-5f) * gamma[c] + beta[c];
    v = fmaxf(v, 0.0f);
    buf[gid] = v * rs_out[r];
}

// ---------------------------------------------------------------- final scale+bias
__global__ void final_out(const float* __restrict__ agg2, const float* __restrict__ rs_in,
                          const float* __restrict__ b2, float* __restrict__ out, int N) {
    long long gid = (long long)blockIdx.x * blockDim.x + threadIdx.x;
    if (gid >= (long long)N * 64) return;
    int c = (int)(gid & 63);
    int r = (int)(gid >> 6);
    out[gid] = agg2[gid] * rs_in[r] + b2[c];
}

// ---------------------------------------------------------------- launcher
extern "C" void kernel_launch(void* const* d_in, const int* in_sizes, int n_in,
                              void* d_out, int out_size, void* d_ws, size_t ws_size,
                              hipStream_t stream) {
    const float* h     = (const float*)d_in[0];
    const float* W1    = (const float*)d_in[1];
    const float* b1    = (const float*)d_in[2];
    const float* W2    = (const float*)d_in[3];
    const float* b2    = (const float*)d_in[4];
    const float* gamma = (const float*)d_in[5];
    const float* beta  = (const float*)d_in[6];
    const int*   src   = (const int*)d_in[7];
    const int*   dst   = (const int*)d_in[8];

    const int N = in_sizes[0] / 128;
    const int E = in_sizes[7];

    // workspace layout (floats): [rs_out N][rs_in N][stats 256][bufB N*128][bufA N*128]
    float* ws     = (float*)d_ws;
    float* rs_out = ws;
    float* rs_in  = ws + N;
    float* stats  = ws + 2LL * N;
    float* bufB   = stats + 256;                 // agg1, then X2 (in place)
    float* bufA   = bufB + (long long)N * 128;   // Y1, then [Y2 | agg2]
    float* Y2     = bufA;
    float* agg2   = bufA + (long long)N * 64;

    const int B = 256;
    const int nblk16 = (N + 15) / 16;

    // zero: rs_out, rs_in, stats, bufB (contiguous from ws start)
    long long zn = 2LL * N + 256 + (long long)N * 128;
    zero_f32<<<(int)((zn + B - 1) / B), B, 0, stream>>>(ws, zn);

    // degrees -> rsqrt(clamp(deg,1))
    count_deg<<<(E + B - 1) / B, B, 0, stream>>>(src, dst, rs_out, rs_in, E);
    finalize_deg<<<(int)((2LL * N + B - 1) / B), B, 0, stream>>>(rs_out, 2LL * N);

    // layer 1: Y1 = (h * rs_out) @ W1   [WMMA f32 16x16x4]
    gemm_wmma_f32<128, true><<<nblk16, 256, 0, stream>>>(h, rs_out, W1, bufA, N);

    // agg1[dst] += Y1[src]
    long long t1 = (long long)E * 32;
    scatter_add<128><<<(int)((t1 + B - 1) / B), B, 0, stream>>>(bufA, src, dst, bufB, E);

    // BN stats over h1 = agg1*rs_in + b1
    bn_stats<<<1024, 128, 0, stream>>>(bufB, rs_in, b1, stats, N);

    // zero agg2 region (second half of bufA) before layer-2 scatter
    zero_f32<<<(int)(((long long)N * 64 + B - 1) / B), B, 0, stream>>>(agg2, (long long)N * 64);

    // BN apply + ReLU + pre-scale by rs_out -> X2 (in place in bufB)
    long long tb = (long long)N * 128;
    bn_apply<<<(int)((tb + B - 1) / B), B, 0, stream>>>(bufB, rs_in, rs_out, b1,
                                                        gamma, beta, stats, N);

    // layer 2: Y2 = X2 @ W2 (X2 already scaled)
    gemm_wmma_f32<64, false><<<nblk16, 128, 0, stream>>>(bufB, nullptr, W2, Y2, N);

    // agg2[dst] += Y2[src]
    long long t2 = (long long)E * 16;
    scatter_add<64><<<(int)((t2 + B - 1) / B), B, 0, stream>>>(Y2, src, dst, agg2, E);

    // out = agg2 * rs_in + b2
    long long to = (long long)N * 64;
    final_out<<<(int)((to + B - 1) / B), B, 0, stream>>>(agg2, rs_in, b2, (float*)d_out, N);
}